// KeyValueMemoryRecallNet_69595650064846
// MI455X (gfx1250) — compile-verified
//
#include <hip/hip_runtime.h>

// ---------------- problem sizes (match reference) ----------------
#define BSZ   2048
#define SEQ   512
#define INSZ  64
#define OUTSZ 16
#define HID   32
#define MEMN  3
#define LANES 32

typedef float v2f __attribute__((ext_vector_type(2)));
typedef float f4  __attribute__((ext_vector_type(4)));
typedef float v8f __attribute__((ext_vector_type(8)));
typedef unsigned int u32x4 __attribute__((ext_vector_type(4)));
typedef int i32x4 __attribute__((ext_vector_type(4)));
typedef int i32x8 __attribute__((ext_vector_type(8)));

#if defined(__has_builtin)
#  if __has_builtin(__builtin_amdgcn_wmma_f32_16x16x4_f32)
#    define HAVE_WMMA 1
#  else
#    define HAVE_WMMA 0
#  endif
#  if __has_builtin(__builtin_amdgcn_tensor_load_to_lds) && __has_builtin(__builtin_amdgcn_s_wait_tensorcnt)
#    define HAVE_TDM 1
#  else
#    define HAVE_TDM 0
#  endif
#else
#  define HAVE_WMMA 0
#  define HAVE_TDM 0
#endif

// ---------------- LDS layout (float offsets), padded strides -----
#define WIH_S 68   // w_ih rows: 66 cols + 2 zero pad
#define WHH_S 34   // 32-col matrices: +2 pad
#define XS    68   // x tile row: 64 in + 2 retr + 2 zero pad (16B aligned rows)
#define HS    34
#define QS    68
#define KS    68
enum : int {
  o_wih  = 0,
  o_whh  = o_wih  + 96*WIH_S,
  o_wq   = o_whh  + 96*WHH_S,
  o_wk   = o_wq   + 64*WHH_S,
  o_wv   = o_wk   + 64*WHH_S,
  o_wo   = o_wv   + 64*WHH_S,
  o_wrec = o_wo   + 16*WHH_S,
  o_brz  = o_wrec + 2*64,
  o_bxn  = o_brz  + 64,
  o_bhn  = o_bxn  + 32,
  o_bq   = o_bhn  + 32,
  o_bk   = o_bq   + 64,
  o_bv   = o_bk   + 64,
  o_bo   = o_bv   + 64,
  o_brec = o_bo   + 16,
  o_x    = o_brec + 4,          // double-buffered input tile
  o_h    = o_x    + 2*16*XS,
  o_q    = o_h    + 16*HS,
  o_keys = o_q    + 16*QS,      // [slot][b][k]
  o_vals = o_keys + MEMN*16*KS,
  TOTF   = o_vals + MEMN*16*KS
};
#define SMEM_BYTES (TOTF * 4)

// ---------------- small helpers ----------------------------------
__device__ __forceinline__ v8f splat8(float s) {
  v8f r;
#pragma unroll
  for (int i = 0; i < 8; ++i) r[i] = s;
  return r;
}

__device__ __forceinline__ v8f sig8(v8f x) {
  v8f r;
#pragma unroll
  for (int i = 0; i < 8; ++i) {
    float e = __builtin_amdgcn_exp2f(-1.4426950408889634f * x[i]);
    r[i] = __builtin_amdgcn_rcpf(1.0f + e);
  }
  return r;
}

__device__ __forceinline__ v8f tanh8(v8f x) {
  v8f r;
#pragma unroll
  for (int i = 0; i < 8; ++i) {
    float e = __builtin_amdgcn_exp2f(2.885390081777927f * x[i]); // exp(2x)
    r[i] = 1.0f - 2.0f * __builtin_amdgcn_rcpf(1.0f + e);
  }
  return r;
}

// D = A(16xK) * W^T(Kx16NT) + D, fp32 WMMA 16x16x4. A,W row-major in LDS.
// A frag: lane&15 = M row, lanes>=16 take K+2 (one ds_load_b64).
// B frag: lane&15 = N col (row n of W), same K split (one ds_load_b64).
template <int NT, int KK>
__device__ __forceinline__ void mmK(v8f* acc, const float* A, int as,
                                    const float* W, int ws, int lane) {
#if HAVE_WMMA
  const int row  = lane & 15;
  const int koff = (lane >> 4) << 1;
#pragma unroll
  for (int kk = 0; kk < KK; ++kk) {
    v2f a = *(const v2f*)(A + row * as + 4 * kk + koff);
#pragma unroll
    for (int j = 0; j < NT; ++j) {
      v2f b = *(const v2f*)(W + (j * 16 + row) * ws + 4 * kk + koff);
      acc[j] = __builtin_amdgcn_wmma_f32_16x16x4_f32(
          false, a, false, b, (short)0, acc[j], false, false);
    }
  }
#else
  const int col = lane & 15, half = lane >> 4;
#pragma unroll
  for (int j = 0; j < NT; ++j)
#pragma unroll
    for (int i = 0; i < 8; ++i) {
      float s = acc[j][i];
      const int M = i + 8 * half, N = j * 16 + col;
      for (int k = 0; k < KK * 4; ++k) s += A[M * as + k] * W[N * ws + k];
      acc[j][i] = s;
    }
#endif
}

// store C-layout accumulator (M=i+8*half, N=lane&15) to LDS row-major
__device__ __forceinline__ void stC(float* dst, int stride, v8f c, int lane) {
  const int col = lane & 15, half = lane >> 4;
#pragma unroll
  for (int i = 0; i < 8; ++i) dst[(i + 8 * half) * stride + col] = c[i];
}

// store C-layout accumulator to global, row stride in floats
__device__ __forceinline__ void stG(float* base, long rstride, v8f c, int lane) {
  const int col = lane & 15, half = lane >> 4;
#pragma unroll
  for (int i = 0; i < 8; ++i) base[(long)(i + 8 * half) * rstride + col] = c[i];
}

__device__ __forceinline__ void cpW(float* dst, const float* src, int rows,
                                    int cols, int stride, int lane) {
  for (int i = lane; i < rows * cols; i += LANES) {
    int r = i / cols, c = i - r * cols;
    dst[r * stride + c] = src[i];
  }
}

// TDM: 2D tile 64x16 f32, batch stride SEQ*INSZ, LDS pad 4 DW per 64 DW row
__device__ __forceinline__ void tdm_issue(unsigned lds_byte_off, const float* g) {
#if HAVE_TDM
  unsigned long long ga = (unsigned long long)(size_t)g;
  u32x4 g0;
  g0[0] = 1u;                                   // count=1 descriptor
  g0[1] = lds_byte_off;                         // lds_addr
  g0[2] = (unsigned)(ga & 0xffffffffull);       // global_addr[31:0]
  g0[3] = (unsigned)((ga >> 32) & 0x1ffffffull) | (2u << 30); // addr[56:32]|type=2
  i32x8 g1;
  g1[0] = (int)((2u << 16) | (1u << 20) | (5u << 22) | (3u << 25));
  //        data_size=4B   pad_en       interval=64DW  amount=4DW
  g1[1] = (int)(64u << 16);  // tensor_dim0 = 64
  g1[2] = (int)(16u << 16);  // tensor_dim1 = 16
  g1[3] = (int)(64u << 16);  // tile_dim0 = 64
  g1[4] = 16;                // tile_dim1 = 16, tile_dim2 = 0
  g1[5] = SEQ * INSZ;        // tensor_dim0_stride (batch row stride, elems)
  g1[6] = 0; g1[7] = 0;
  i32x4 z4 = {0, 0, 0, 0};
#if __clang_major__ >= 23
  i32x8 z8 = {0, 0, 0, 0, 0, 0, 0, 0};
  __builtin_amdgcn_tensor_load_to_lds(g0, g1, z4, z4, z8, 0);
#else
  __builtin_amdgcn_tensor_load_to_lds(g0, g1, z4, z4, 0);
#endif
#endif
}

// ---------------- kernel: one wave per 16-batch tile --------------
__global__ __launch_bounds__(32) void KeyValueMemoryRecallNet_69595650064846_kernel(
    const float* __restrict__ inputs, const float* __restrict__ w_ih,
    const float* __restrict__ w_hh, const float* __restrict__ b_ih,
    const float* __restrict__ b_hh, const float* __restrict__ W_key,
    const float* __restrict__ b_key, const float* __restrict__ W_query,
    const float* __restrict__ b_query, const float* __restrict__ W_value,
    const float* __restrict__ b_value, const float* __restrict__ W_recall,
    const float* __restrict__ b_recall, const float* __restrict__ W_out,
    const float* __restrict__ b_out, float* __restrict__ outp,
    float* __restrict__ hidp) {
  extern __shared__ float sm[];
  const int lane = threadIdx.x;
  const int b0 = blockIdx.x * 16;

  // zero all of LDS (keys/vals/x pads/retr must start at 0)
  for (int i = lane; i < TOTF; i += LANES) sm[i] = 0.0f;
  __syncthreads();

  // stage weights into padded LDS
  cpW(sm + o_wih, w_ih, 96, 66, WIH_S, lane);
  cpW(sm + o_whh, w_hh, 96, 32, WHH_S, lane);
  cpW(sm + o_wq, W_query, 64, 32, WHH_S, lane);
  cpW(sm + o_wk, W_key, 64, 32, WHH_S, lane);
  cpW(sm + o_wv, W_value, 64, 32, WHH_S, lane);
  cpW(sm + o_wo, W_out, 16, 32, WHH_S, lane);
  cpW(sm + o_wrec, W_recall, 2, 64, 64, lane);
  for (int i = lane; i < 64; i += LANES) sm[o_brz + i] = b_ih[i] + b_hh[i];
  for (int i = lane; i < 32; i += LANES) {
    sm[o_bxn + i] = b_ih[64 + i];
    sm[o_bhn + i] = b_hh[64 + i];
  }
  for (int i = lane; i < 64; i += LANES) {
    sm[o_bq + i] = b_query[i];
    sm[o_bk + i] = b_key[i];
    sm[o_bv + i] = b_value[i];
  }
  for (int i = lane; i < 16; i += LANES) sm[o_bo + i] = b_out[i];
  if (lane < 2) sm[o_brec + lane] = b_recall[lane];
  __syncthreads();

  // per-lane bias values (column = lane&15, same for both lane halves)
  const int col = lane & 15;
  float brz_r[4], bq_r[4], bk_r[4], bv_r[4], bxn_r[2], bhn_r[2], bo_r;
#pragma unroll
  for (int j = 0; j < 4; ++j) {
    brz_r[j] = sm[o_brz + j * 16 + col];
    bq_r[j] = sm[o_bq + j * 16 + col];
    bk_r[j] = sm[o_bk + j * 16 + col];
    bv_r[j] = sm[o_bv + j * 16 + col];
  }
#pragma unroll
  for (int j = 0; j < 2; ++j) {
    bxn_r[j] = sm[o_bxn + j * 16 + col];
    bhn_r[j] = sm[o_bhn + j * 16 + col];
  }
  bo_r = sm[o_bo + col];

  v8f h0 = splat8(0.0f), h1 = splat8(0.0f);  // hidden state, C layout

#if HAVE_TDM
  const unsigned ldsx0 =
      __builtin_amdgcn_readfirstlane((unsigned)(size_t)(sm + o_x));
  const unsigned ldsx1 = ldsx0 + 16 * XS * 4;
  tdm_issue(ldsx0, inputs + (size_t)b0 * SEQ * INSZ);  // prefetch t=0
#endif

  for (int t = 0; t < SEQ; ++t) {
#if HAVE_TDM
    __builtin_amdgcn_s_wait_tensorcnt(0);  // x(t) tile resident in LDS
    if (t + 1 < SEQ)
      tdm_issue(((t + 1) & 1) ? ldsx1 : ldsx0,
                inputs + ((size_t)b0 * SEQ + (t + 1)) * INSZ);
#else
    {  // synchronous fallback load of x(t): 2 lanes per batch row
      const int r = lane >> 1, hf = lane & 1;
      const float* g = inputs + ((size_t)(b0 + r) * SEQ + t) * INSZ + hf * 32;
      float* xd = sm + o_x + (t & 1) * (16 * XS) + r * XS + hf * 32;
#pragma unroll
      for (int j = 0; j < 8; ++j) ((f4*)xd)[j] = ((const f4*)g)[j];
    }
    __syncthreads();
#endif

    // ---- GRU gates via fp32 WMMA ----
    const float* xb = sm + o_x + (t & 1) * (16 * XS);
    v8f crz[4], cxn[2], chn[2];
#pragma unroll
    for (int j = 0; j < 4; ++j) crz[j] = splat8(brz_r[j]);
#pragma unroll
    for (int j = 0; j < 2; ++j) {
      cxn[j] = splat8(bxn_r[j]);
      chn[j] = splat8(bhn_r[j]);
    }
    mmK<4, 17>(crz, xb, XS, sm + o_wih, WIH_S, lane);            // x @ w_ih[rz]
    mmK<2, 17>(cxn, xb, XS, sm + o_wih + 64 * WIH_S, WIH_S, lane); // x @ w_ih[n]
    mmK<4, 8>(crz, sm + o_h, HS, sm + o_whh, WHH_S, lane);        // h @ w_hh[rz]
    mmK<2, 8>(chn, sm + o_h, HS, sm + o_whh + 64 * WHH_S, WHH_S, lane);

    v8f r0 = sig8(crz[0]), r1 = sig8(crz[1]);
    v8f z0 = sig8(crz[2]), z1 = sig8(crz[3]);
    v8f n0 = tanh8(cxn[0] + r0 * chn[0]);
    v8f n1 = tanh8(cxn[1] + r1 * chn[1]);
    h0 = n0 + z0 * (h0 - n0);
    h1 = n1 + z1 * (h1 - n1);
    if (t == MEMN) { h0 = splat8(0.0f); h1 = splat8(0.0f); }  // flush

    stC(sm + o_h, HS, h0, lane);
    stC(sm + o_h + 16, HS, h1, lane);
    {  // hiddens output (B,S,32)
      float* hb = hidp + (size_t)b0 * SEQ * HID + (size_t)t * HID;
      stG(hb, (long)SEQ * HID, h0, lane);
      stG(hb + 16, (long)SEQ * HID, h1, lane);
    }
    __syncthreads();  // h_new visible for A-operand reads

    // ---- projections off h_new ----
    v8f cq[4];
#pragma unroll
    for (int j = 0; j < 4; ++j) cq[j] = splat8(bq_r[j]);
    mmK<4, 8>(cq, sm + o_h, HS, sm + o_wq, WHH_S, lane);
#pragma unroll
    for (int j = 0; j < 4; ++j) stC(sm + o_q + j * 16, QS, cq[j], lane);

    if (t < MEMN) {  // write memory slot t (slots start zeroed -> assignment)
      v8f ck[4], cv[4];
#pragma unroll
      for (int j = 0; j < 4; ++j) { ck[j] = splat8(bk_r[j]); cv[j] = splat8(bv_r[j]); }
      mmK<4, 8>(ck, sm + o_h, HS, sm + o_wk, WHH_S, lane);
      mmK<4, 8>(cv, sm + o_h, HS, sm + o_wv, WHH_S, lane);
#pragma unroll
      for (int j = 0; j < 4; ++j) {
        stC(sm + o_keys + t * 16 * KS + j * 16, KS, ck[j], lane);
        stC(sm + o_vals + t * 16 * KS + j * 16, KS, cv[j], lane);
      }
    }

    v8f co[1];
    co[0] = splat8(bo_r);
    mmK<1, 8>(co, sm + o_h, HS, sm + o_wo, WHH_S, lane);
    stG(outp + (size_t)b0 * SEQ * OUTSZ + (size_t)t * OUTSZ,
        (long)SEQ * OUTSZ, co[0], lane);
    __syncthreads();  // q / keys / vals visible for attention

    // ---- attention + recall (VALU, 2 lanes per batch row) ----
    {
      const int bb = lane >> 1, h2 = lane & 1;
      const float* qp = sm + o_q + bb * QS + h2 * 32;
      f4 qv[8];
#pragma unroll
      for (int j = 0; j < 8; ++j) qv[j] = *(const f4*)(qp + 4 * j);
      float sc[3];
#pragma unroll
      for (int m = 0; m < MEMN; ++m) {
        const float* kp = sm + o_keys + (m * 16 + bb) * KS + h2 * 32;
        float s = 0.0f;
#pragma unroll
        for (int j = 0; j < 8; ++j) {
          f4 kv = *(const f4*)(kp + 4 * j);
          s += qv[j][0] * kv[0] + qv[j][1] * kv[1] + qv[j][2] * kv[2] +
               qv[j][3] * kv[3];
        }
        s += __shfl_xor(s, 1, 32);  // combine the two half-rows
        sc[m] = s;
      }
      const int counter = t < MEMN ? t : MEMN;  // slots written so far
      float mx = -3.0e38f;
      for (int m = 0; m < counter; ++m) mx = fmaxf(mx, sc[m]);
      float e[3] = {0.0f, 0.0f, 0.0f}, ssum = 0.0f;
      for (int m = 0; m < counter; ++m) {
        e[m] = __builtin_amdgcn_exp2f((sc[m] - mx) * 1.4426950408889634f);
        ssum += e[m];
      }
      const float inv = counter > 0 ? __builtin_amdgcn_rcpf(ssum) : 0.0f;
      f4 rv[8];
      const f4 fz = {0.0f, 0.0f, 0.0f, 0.0f};
#pragma unroll
      for (int j = 0; j < 8; ++j) rv[j] = fz;
      for (int m = 0; m < counter; ++m) {
        const float am = e[m] * inv;
        const float* vp = sm + o_vals + (m * 16 + bb) * KS + h2 * 32;
#pragma unroll
        for (int j = 0; j < 8; ++j) rv[j] += am * *(const f4*)(vp + 4 * j);
      }
      float pr[2];
#pragma unroll
      for (int jj = 0; jj < 2; ++jj) {
        const float* wp = sm + o_wrec + jj * 64 + h2 * 32;
        float s = 0.0f;
#pragma unroll
        for (int j = 0; j < 8; ++j) {
          f4 wv = *(const f4*)(wp + 4 * j);
          s += rv[j][0] * wv[0] + rv[j][1] * wv[1] + rv[j][2] * wv[2] +
               rv[j][3] * wv[3];
        }
        s += __shfl_xor(s, 1, 32);
        pr[jj] = (counter > 0) ? (s + sm[o_brec + jj]) : 0.0f;
      }
      // retr(t) feeds x columns 64..65 of the NEXT buffer (TDM never touches them)
      if (h2 == 0 && t + 1 < SEQ) {
        float* xr = sm + o_x + ((t + 1) & 1) * (16 * XS) + bb * XS + 64;
        xr[0] = pr[0];
        xr[1] = pr[1];
      }
    }
    __syncthreads();  // retr visible before next step's gx
  }
}

extern "C" void kernel_launch(void* const* d_in, const int* in_sizes, int n_in,
                              void* d_out, int out_size, void* d_ws,
                              size_t ws_size, hipStream_t stream) {
  const float* inputs = (const float*)d_in[0];
  const float* w_ih = (const float*)d_in[1];
  const float* w_hh = (const float*)d_in[2];
  const float* b_ih = (const float*)d_in[3];
  const float* b_hh = (const float*)d_in[4];
  const float* W_key = (const float*)d_in[5];
  const float* b_key = (const float*)d_in[6];
  const float* W_query = (const float*)d_in[7];
  const float* b_query = (const float*)d_in[8];
  const float* W_value = (const float*)d_in[9];
  const float* b_value = (const float*)d_in[10];
  const float* W_recall = (const float*)d_in[11];
  const float* b_recall = (const float*)d_in[12];
  const float* W_out = (const float*)d_in[13];
  const float* b_out = (const float*)d_in[14];
  float* outp = (float*)d_out;
  float* hidp = outp + (size_t)BSZ * SEQ * OUTSZ;

  (void)hipFuncSetAttribute(
      (const void*)KeyValueMemoryRecallNet_69595650064846_kernel,
      hipFuncAttributeMaxDynamicSharedMemorySize, SMEM_BYTES);
  KeyValueMemoryRecallNet_69595650064846_kernel<<<dim3(BSZ / 16), dim3(32),
                                                  SMEM_BYTES, stream>>>(
      inputs, w_ih, w_hh, b_ih, b_hh, W_key, b_key, W_query, b_query, W_value,
      b_value, W_recall, b_recall, W_out, b_out, outp, hidp);
}